// MCPBRNN_Generic_constantoutput_variableLoss_MCA2_10196252360844
// MI455X (gfx1250) — compile-verified
//
#include <hip/hip_runtime.h>
#include <math.h>

#define SPIN_LEN 365
#define TRAIN_LEN 200000
#define ML_C 2.9086f
#define SL_C 1.898f

#define THREADS 256
#define EPT 16                    // elements per thread (contiguous, preserves scan order)
#define CHUNK (THREADS * EPT)     // 4096 elements per block
#define STD_BLOCKS 64
#define ROW_F2 17                 // padded LDS row pitch in float2 (136B) -> bank-conflict-free

// ---------------------------------------------------------------------------
// gate scalars from the 6 weight inputs (cheap; recomputed per thread)
// ---------------------------------------------------------------------------
__device__ __forceinline__ void load_gates(const float* w_om, const float* w_gw,
                                           const float* w_lm, const float* w_fm,
                                           const float* b0p,  const float* wb2p,
                                           float& oo, float& oogw, float& ol1,
                                           float& b0, float& wb2) {
    float e_om = expf(w_om[0]);
    float e_gw = expf(w_gw[0]);
    float e_lm = expf(w_lm[0]);
    float e_fm = expf(w_fm[0]);
    float denom = e_om + e_gw + e_lm + e_fm;
    oo   = e_om / denom;
    oogw = e_gw / denom;
    ol1  = e_lm / denom;
    b0   = b0p[0];
    wb2  = wb2p[0];
}

__device__ __forceinline__ void elem_fol(float u2, float oo, float oogw, float ol1,
                                         float b0, float wb2, float& f, float& ol) {
    float z  = b0 + (u2 - ML_C) / SL_C * wb2;
    float sg = 1.0f / (1.0f + expf(-z));
    ol = ol1 * sg;
    f  = 1.0f - oo - ol - oogw;     // same association order as the reference
}

// ---------------------------------------------------------------------------
// Stage one block tile (CHUNK float2) into LDS with the CDNA5 async
// VMEM->LDS engine. Lane-interleaved global order (fully coalesced); LDS
// layout is per-thread rows of 16 float2 with 1-float2 pad (conflict-free).
// ---------------------------------------------------------------------------
__device__ __forceinline__ void stage_tile(const float2* __restrict__ xp,
                                           long long blockBase, int tl, long long Bp,
                                           float2* tile /* __shared__ */) {
    int tid = threadIdx.x;
    uint32_t tbase = (uint32_t)(uintptr_t)tile;   // low 32 bits = LDS byte offset
#pragma unroll
    for (int k = 0; k < EPT; ++k) {
        int m = tid + k * THREADS;                // block-local element index
        long long j = blockBase + m;
        if (j < Bp) {
            int row = m >> 4;
            int e   = m & 15;
            uint32_t dst = tbase + (uint32_t)(row * ROW_F2 + e) * 8u;
            const float2* src = xp + (j + tl);
            asm volatile("global_load_async_to_lds_b64 %0, %1, off"
                         :: "v"(dst), "v"(src) : "memory");
        }
    }
    asm volatile("s_wait_asynccnt 0x0" ::: "memory");
    __syncthreads();
}

// ordered inclusive Hillis-Steele scan of affine pairs in LDS
__device__ __forceinline__ void lds_affine_scan(float2* sh, int tid) {
    for (int s = 1; s < THREADS; s <<= 1) {
        float2 v = sh[tid];
        if (tid >= s) {
            float2 L = sh[tid - s];
            v = make_float2(L.x * v.x, L.y * v.x + v.y);
        }
        __syncthreads();
        sh[tid] = v;
        __syncthreads();
    }
}

// ---------------------------------------------------------------------------
// Phase 1: blocks [0,G)   -> per-block affine aggregate (a,b) of c' = f*c + u1
//          blocks [G,G+64)-> partial (sum, sumsq) for std(y[365:200000], ddof=1)
// ---------------------------------------------------------------------------
__global__ void __launch_bounds__(THREADS)
k_phase1(const float* __restrict__ x, const float* __restrict__ y,
         const int* __restrict__ tlp,
         const float* w_om, const float* w_gw, const float* w_lm, const float* w_fm,
         const float* b0p, const float* wb2p,
         float2* __restrict__ blockAgg, double* __restrict__ partial,
         int B, int G) {
    __shared__ float2 tile[THREADS * ROW_F2];
    __shared__ float2 shs[THREADS];
    __shared__ double shd[2 * THREADS];

    int tid = threadIdx.x;

    if ((int)blockIdx.x >= G) {
        // ---- std partial path ----
        int sb = (int)blockIdx.x - G;
        int n1 = (B < TRAIN_LEN) ? B : TRAIN_LEN;
        double s = 0.0, ss = 0.0;
        for (int i = SPIN_LEN + sb * THREADS + tid; i < n1; i += STD_BLOCKS * THREADS) {
            double v = (double)y[i];
            s += v; ss += v * v;
        }
        shd[tid] = s; shd[THREADS + tid] = ss;
        __syncthreads();
        for (int st = THREADS / 2; st > 0; st >>= 1) {
            if (tid < st) {
                shd[tid]           += shd[tid + st];
                shd[THREADS + tid] += shd[THREADS + tid + st];
            }
            __syncthreads();
        }
        if (tid == 0) { partial[2 * sb] = shd[0]; partial[2 * sb + 1] = shd[THREADS]; }
        return;
    }

    // ---- scan aggregate path ----
    int tl = tlp[0];
    long long Bp = (long long)B - tl;
    float oo, oogw, ol1, b0, wb2;
    load_gates(w_om, w_gw, w_lm, w_fm, b0p, wb2p, oo, oogw, ol1, b0, wb2);

    long long blockBase = (long long)blockIdx.x * CHUNK;
    stage_tile((const float2*)x, blockBase, tl, Bp, tile);

    const float2* myrow = &tile[tid * ROW_F2];
    long long tbase = blockBase + (long long)tid * EPT;
    float a = 1.0f, b = 0.0f;
#pragma unroll
    for (int e = 0; e < EPT; ++e) {
        long long j = tbase + e;
        if (j < Bp) {
            float2 u = myrow[e];
            float f, ol;
            elem_fol(u.y, oo, oogw, ol1, b0, wb2, f, ol);
            b = b * f + u.x;
            a = a * f;
        }
    }
    shs[tid] = make_float2(a, b);
    __syncthreads();
    lds_affine_scan(shs, tid);
    if (tid == THREADS - 1) blockAgg[blockIdx.x] = shs[tid];
}

// ---------------------------------------------------------------------------
// Phase 2: every block redundantly (a) finalizes obsstd from the 64 partials,
// (b) scans the G block aggregates in LDS to get its own exclusive prefix,
// then does the intra-block thread scan and writes all 13 output vectors.
// ---------------------------------------------------------------------------
__global__ void __launch_bounds__(THREADS)
k_phase2(const float* __restrict__ x, const int* __restrict__ tlp,
         const float* w_om, const float* w_gw, const float* w_lm, const float* w_fm,
         const float* b0p, const float* wb2p,
         const float2* __restrict__ blockAgg, const double* __restrict__ partial,
         float* __restrict__ out, int B, int G) {
    __shared__ float2 tile[THREADS * ROW_F2];
    __shared__ float2 shs[THREADS];
    __shared__ double shd[2 * STD_BLOCKS];
    __shared__ float  s_std;
    __shared__ float2 s_bpref;

    int tid = threadIdx.x;
    int tl  = tlp[0];
    long long Bp = (long long)B - tl;

    // ---- finalize obsstd (redundant per block; 64 doubles) ----
    if (tid < STD_BLOCKS) {
        shd[tid]              = partial[2 * tid];
        shd[STD_BLOCKS + tid] = partial[2 * tid + 1];
    }
    __syncthreads();
    for (int st = STD_BLOCKS / 2; st > 0; st >>= 1) {
        if (tid < st) {
            shd[tid]              += shd[tid + st];
            shd[STD_BLOCKS + tid] += shd[STD_BLOCKS + tid + st];
        }
        __syncthreads();
    }
    if (tid == 0) {
        int n1 = (B < TRAIN_LEN) ? B : TRAIN_LEN;
        double n = (double)(n1 - SPIN_LEN);
        double sum = shd[0], sumsq = shd[STD_BLOCKS];
        double stdv = 0.0;
        if (n > 1.5) {
            double var = (sumsq - sum * sum / n) / (n - 1.0);
            stdv = sqrt(var > 0.0 ? var : 0.0);
        }
        s_std = (float)stdv;
    }

    // ---- grid-level exclusive prefix for this block (ordered LDS scan) ----
    shs[tid] = (tid < G) ? blockAgg[tid] : make_float2(1.0f, 0.0f);
    __syncthreads();
    lds_affine_scan(shs, tid);
    if (tid == 0)
        s_bpref = (blockIdx.x == 0) ? make_float2(1.0f, 0.0f) : shs[blockIdx.x - 1];
    __syncthreads();

    // ---- stage tile + per-thread element values ----
    float oo, oogw, ol1, b0, wb2;
    load_gates(w_om, w_gw, w_lm, w_fm, b0p, wb2p, oo, oogw, ol1, b0, wb2);

    long long blockBase = (long long)blockIdx.x * CHUNK;
    stage_tile((const float2*)x, blockBase, tl, Bp, tile);

    const float2* myrow = &tile[tid * ROW_F2];
    long long tbase = blockBase + (long long)tid * EPT;

    float fv[EPT], olv[EPT], u1v[EPT];
    float a = 1.0f, b = 0.0f;
#pragma unroll
    for (int e = 0; e < EPT; ++e) {
        long long j = tbase + e;
        float f = 1.0f, ol = 0.0f, u1 = 0.0f;
        if (j < Bp) {
            float2 u = myrow[e];
            elem_fol(u.y, oo, oogw, ol1, b0, wb2, f, ol);
            u1 = u.x;
            b = b * f + u1;
            a = a * f;
        }
        fv[e] = f; olv[e] = ol; u1v[e] = u1;
    }

    // ---- intra-block exclusive scan of thread aggregates ----
    shs[tid] = make_float2(a, b);
    __syncthreads();
    lds_affine_scan(shs, tid);

    float2 bp  = s_bpref;
    float c0bl = bp.y;                                   // c at block start (c_init = 0)
    float2 ex  = (tid == 0) ? make_float2(1.0f, 0.0f) : shs[tid - 1];
    float c    = ex.x * c0bl + ex.y;                     // carry entering this thread's chunk
    float obsstd = s_std;

    long long Bll = (long long)B;
    float* h_n   = out;
    float* c_n   = out + 1 * Bll;
    float* l_n   = out + 2 * Bll;
    float* gw_n  = out + 3 * Bll;
    float* bp_n  = out + 4 * Bll;
    float* g_ib  = out + 5 * Bll;
    float* g_oo  = out + 6 * Bll;
    float* g_gw  = out + 7 * Bll;
    float* g_ol  = out + 8 * Bll;
    float* g_f   = out + 9 * Bll;
    float* hout  = out + 10 * Bll;                       // (B,2) row-major
    float* ostd  = out + 12 * Bll;

#pragma unroll
    for (int e = 0; e < EPT; ++e) {
        long long j = tbase + e;
        if (j < Bp) {
            long long i = j + tl;
            float cp = c;                                // c_prev at step j
            float ol = olv[e], f = fv[e];
            float h  = oo * cp;
            h_n[i]  = h;
            c_n[i]  = cp;
            l_n[i]  = ol * cp;
            gw_n[i] = oogw * cp;
            bp_n[i] = 0.0f;
            g_ib[i] = 0.0f;
            g_oo[i] = oo;
            g_gw[i] = oogw;
            g_ol[i] = ol;
            g_f[i]  = f;
            hout[2 * i]     = h;
            hout[2 * i + 1] = obsstd;
            ostd[i] = obsstd;
            c = f * cp + u1v[e];
        }
    }

    if (blockIdx.x == 0) {                               // front padding rows (time_lag > 0)
        for (long long i = tid; i < (long long)tl; i += blockDim.x) {
            h_n[i] = 0.0f; c_n[i] = 0.0f; l_n[i] = 0.0f; gw_n[i] = 0.0f;
            bp_n[i] = 0.0f; g_ib[i] = 0.0f; g_oo[i] = 0.0f; g_gw[i] = 0.0f;
            g_ol[i] = 0.0f; g_f[i] = 0.0f;
            hout[2 * i] = 0.0f; hout[2 * i + 1] = 0.0f; ostd[i] = 0.0f;
        }
    }
}

// ---------------------------------------------------------------------------
// Host side: two launches only (workload is launch-latency bound)
// ---------------------------------------------------------------------------
extern "C" void kernel_launch(void* const* d_in, const int* in_sizes, int n_in,
                              void* d_out, int out_size, void* d_ws, size_t ws_size,
                              hipStream_t stream) {
    const float* x    = (const float*)d_in[0];
    const float* yobs = (const float*)d_in[1];
    // d_in[2] = epoch (unused)
    const int*   tlp  = (const int*)d_in[3];
    const float* w_om = (const float*)d_in[4];
    const float* w_gw = (const float*)d_in[5];
    const float* w_lm = (const float*)d_in[6];
    const float* w_fm = (const float*)d_in[7];
    const float* b0p  = (const float*)d_in[8];
    const float* wb2p = (const float*)d_in[9];
    float* out = (float*)d_out;

    int B = in_sizes[1];                       // y_obs is (B,1); S == 1
    int G = (B + CHUNK - 1) / CHUNK;           // 64 blocks for B = 262144

    // workspace layout
    double* partial  = (double*)d_ws;                                   // 2*STD_BLOCKS doubles
    float2* blockAgg = (float2*)((char*)d_ws + 2 * STD_BLOCKS * sizeof(double));  // G float2

    k_phase1<<<G + STD_BLOCKS, THREADS, 0, stream>>>(x, yobs, tlp,
                                                     w_om, w_gw, w_lm, w_fm, b0p, wb2p,
                                                     blockAgg, partial, B, G);
    k_phase2<<<G, THREADS, 0, stream>>>(x, tlp,
                                        w_om, w_gw, w_lm, w_fm, b0p, wb2p,
                                        blockAgg, partial, out, B, G);
}